// MLPdirected_59519656788458
// MI455X (gfx1250) — compile-verified
//
#include <hip/hip_runtime.h>
#include <hip/hip_bf16.h>

#define B_ 2
#define F_ 32
#define N_ 1024
#define H_ 64
#define LDSTRIDE 68   // 64 + 4 dword pad: keeps v4f alignment, kills bank conflicts

typedef __attribute__((ext_vector_type(2))) float v2f;
typedef __attribute__((ext_vector_type(4))) float v4f;
typedef __attribute__((ext_vector_type(8))) float v8f;

// Kernel 1: hi'[b,n,h] = sum_f e[b,n,f]*W1[h,f] + b1[h] ; hj[b,n,h] = sum_f e*W1[h,F+f]
// emb is (B, F, N) so e[b,n,f] = emb[(b*F+f)*N + n]. One block per (b,n), 64 threads (h).
__global__ void node_hidden_kernel(const float* __restrict__ emb,
                                   const float* __restrict__ W1,
                                   const float* __restrict__ b1,
                                   float* __restrict__ hi,
                                   float* __restrict__ hj) {
  __shared__ float e[F_];
  int bn = blockIdx.x;
  int b = bn >> 10;             // N_ == 1024
  int n = bn & (N_ - 1);
  int h = threadIdx.x;          // 0..63
  if (h < F_) e[h] = emb[(b * F_ + h) * N_ + n];
  __syncthreads();
  const float* w = W1 + h * (2 * F_);
  float ai = 0.f, aj = 0.f;
#pragma unroll
  for (int f = 0; f < F_; ++f) {
    float ev = e[f];
    ai = fmaf(ev, w[f], ai);
    aj = fmaf(ev, w[F_ + f], aj);
  }
  hi[bn * H_ + h] = ai + b1[h];   // fold b1 into hi
  hj[bn * H_ + h] = aj;
}

// Kernel 2: out[b,i,j] = sigmoid( sum_h relu(hi[i,h]+hj[j,h]) * w2[h] + b2 )
// Wave computes a 16(i) x 16(j) tile. Contraction over h via V_WMMA_F32_16X16X4_F32:
//   A[m,k] = relu(hi[i, kb*4+k] + hj[j0+m, kb*4+k]), B[k,n] = w2[kb*4+k] (replicated cols).
__global__ void edge_mlp_kernel(const float* __restrict__ hi,
                                const float* __restrict__ hj,
                                const float* __restrict__ W2,
                                const float* __restrict__ b2,
                                float* __restrict__ out) {
  __shared__ float hjs[16 * LDSTRIDE];
  int blk = blockIdx.x;          // 1024 blocks
  int jt  = blk & 63;            // N/16 = 64 j-tiles
  int tmp = blk >> 6;
  int b   = tmp >> 3;            // batch
  int itg = tmp & 7;             // group of 8 i-tiles (one per wave)

  // Cooperatively stage the 16x64 hj tile into LDS (256 threads x 1 v4f each).
  {
    int t = threadIdx.x;               // 0..255 -> 1024 dwords
    int row = t >> 4;                  // 0..15
    int c4  = t & 15;                  // 0..15 (v4f column)
    const v4f* src = reinterpret_cast<const v4f*>(hj + (size_t)(b * N_ + jt * 16) * H_);
    *reinterpret_cast<v4f*>(&hjs[row * LDSTRIDE + c4 * 4]) = src[t];
  }
  __syncthreads();

  int lane = threadIdx.x & 31;
  int wave = threadIdx.x >> 5;
  int it   = itg * 8 + wave;           // this wave's i-tile
  int m    = lane & 15;                // A-matrix row (j within tile)
  int k0   = (lane >> 4) * 2;          // A-matrix K pair: lanes 0-15 -> K0,1; 16-31 -> K2,3

  // Preload the 16 B-fragments of w2 (per lane: {w2[4kb+k0], w2[4kb+k0+1]}).
  v2f wfrag[16];
#pragma unroll
  for (int kb = 0; kb < 16; ++kb)
    wfrag[kb] = *reinterpret_cast<const v2f*>(W2 + kb * 4 + k0);
  float b2v = b2[0];

  for (int il = 0; il < 16; ++il) {
    int i = it * 16 + il;
    const float* hirow = hi + (size_t)(b * N_ + i) * H_;
    v8f acc = {};
#pragma unroll
    for (int kb = 0; kb < 16; ++kb) {
      int h = kb * 4 + k0;
      v2f hv = *reinterpret_cast<const v2f*>(hirow + h);             // broadcast load
      v2f jv = *reinterpret_cast<const v2f*>(&hjs[m * LDSTRIDE + h]); // ds_load_b64
      v2f a;
      a.x = fmaxf(hv.x + jv.x, 0.f);
      a.y = fmaxf(hv.y + jv.y, 0.f);
      // D(16x16,f32) += A(16x4,f32) x B(4x16,f32)
      acc = __builtin_amdgcn_wmma_f32_16x16x4_f32(false, a, false, wfrag[kb],
                                                  (short)0, acc, false, false);
    }
    // D rows 0-7 live in acc[0..7] of lanes 0-15 (all 16 cols identical);
    // rows 8-15 in acc[0..7] of lanes 16-31. Lanes 0 and 16 commit the 16 outputs.
    if ((lane & 15) == 0) {
      float* op = out + ((size_t)(b * N_ + i) * N_) + jt * 16 + (lane >> 4) * 8;
      v4f o0, o1;
#pragma unroll
      for (int r = 0; r < 4; ++r) {
        o0[r] = 1.0f / (1.0f + __expf(-(acc[r]     + b2v)));
        o1[r] = 1.0f / (1.0f + __expf(-(acc[r + 4] + b2v)));
      }
      *reinterpret_cast<v4f*>(op)     = o0;
      *reinterpret_cast<v4f*>(op + 4) = o1;
    }
  }
}

extern "C" void kernel_launch(void* const* d_in, const int* in_sizes, int n_in,
                              void* d_out, int out_size, void* d_ws, size_t ws_size,
                              hipStream_t stream) {
  // inputs: adj_in(0, unused), emb_in(1), layer(2, unused), W1(3), b1(4), W2(5), b2(6)
  const float* emb = (const float*)d_in[1];
  const float* W1  = (const float*)d_in[3];
  const float* b1  = (const float*)d_in[4];
  const float* W2  = (const float*)d_in[5];
  const float* b2  = (const float*)d_in[6];
  float* out = (float*)d_out;

  float* hi = (float*)d_ws;                  // B*N*H floats (b1 folded in)
  float* hj = hi + (size_t)B_ * N_ * H_;     // B*N*H floats  (total 1 MB of d_ws)

  node_hidden_kernel<<<B_ * N_, 64, 0, stream>>>(emb, W1, b1, hi, hj);
  edge_mlp_kernel<<<1024, 256, 0, stream>>>(hi, hj, W2, b2, out);
}